// HippoLegsCell_65171833749527
// MI455X (gfx1250) — compile-verified
//
#include <hip/hip_runtime.h>

// HiPPO-LegS cell update on gfx1250:
//   out[b, m] = sum_n c_t[b, n] * A[t][m, n]  +  B[t][m] * input[b]
// Batched skinny GEMM (524288 x 64) @ (64 x 64)^T via V_WMMA_F32_16X16X4_F32.
// Memory-bound (~258 MB @ 23.3 TB/s ~= 11 us), so full-precision f32 WMMA is
// the right choice: matches reference numerics, matrix pipe easily keeps up.

typedef float v2f __attribute__((ext_vector_type(2)));
typedef float v4f __attribute__((ext_vector_type(4)));
typedef float v8f __attribute__((ext_vector_type(8)));

// Row stride (in floats) for LDS tiles: 64 + 4 pad so that column-strided
// fragment reads (fixed col, rows 0..15 across lanes) are bank-conflict-free:
// bank = (row*68 + col) % 64 = (4*row + col) % 64 -> 16 distinct banks per
// half-wave, halves offset by +2 cols -> disjoint. b64 reads cover all 64 banks.
#define LDS_STRIDE 68

__global__ __launch_bounds__(256) void hippo_legs_wmma_kernel(
    const float* __restrict__ inp,    // (BATCH)        input[b]
    const float* __restrict__ c_t,    // (BATCH, 64)
    const float* __restrict__ Afull,  // (1024, 64, 64)
    const float* __restrict__ Bfull,  // (1024, 64)
    const int*   __restrict__ t_ptr,  // scalar t
    float*       __restrict__ out)    // (BATCH, 64)
{
    __shared__ float sAt[64 * LDS_STRIDE];       // A[t], 17408 B
    __shared__ float sC[8 * 16 * LDS_STRIDE];    // per-wave c_t tiles, 34816 B

    const int t = *t_ptr;
    const float* __restrict__ At = Afull + (size_t)t * 64 * 64;
    const float* __restrict__ Bt = Bfull + (size_t)t * 64;

    const int tid  = threadIdx.x;     // 0..255 (8 waves of 32)
    const int wave = tid >> 5;
    const int lane = tid & 31;
    const int half = lane >> 4;       // 0 | 1
    const int l16  = lane & 15;

    const long long b0 = (long long)blockIdx.x * 128 + (long long)wave * 16;

    // ---- Block-cooperative load of A[t] (64x64 f32) into padded LDS ----
    // 1024 float4s, 256 threads x 4 iterations, fully coalesced.
    #pragma unroll
    for (int i = 0; i < 4; ++i) {
        int q    = tid + 256 * i;       // float4 id, 0..1023
        int row  = q >> 4;              // 16 float4 per 64-float row
        int col4 = q & 15;
        v4f v = *(const v4f*)(At + row * 64 + col4 * 4);
        *(v4f*)(&sAt[row * LDS_STRIDE + col4 * 4]) = v;
    }

    // ---- Wave loads its 16x64 c_t tile into padded LDS (coalesced) ----
    {
        float* sc = sC + wave * 16 * LDS_STRIDE;
        #pragma unroll
        for (int i = 0; i < 8; ++i) {
            int q    = lane + 32 * i;   // float4 id, 0..255
            int row  = q >> 4;
            int col4 = q & 15;
            v4f v = *(const v4f*)(c_t + (b0 + row) * 64 + col4 * 4);
            *(v4f*)(&sc[row * LDS_STRIDE + col4 * 4]) = v;
        }
    }

    __syncthreads();

    // ---- A-operand fragments (held in regs across all 4 N-tiles) ----
    // ISA f32 A 16x4 layout: lane%16 = M row, K = vgpr + 2*(lane/16).
    // afrag[kc] = c_t[b0 + l16, 4*kc + 2*half + {0,1}] -> one b64 LDS read each.
    v2f afrag[16];
    {
        const float* sc = sC + wave * 16 * LDS_STRIDE + l16 * LDS_STRIDE + 2 * half;
        #pragma unroll
        for (int kc = 0; kc < 16; ++kc)
            afrag[kc] = *(const v2f*)(sc + 4 * kc);
    }

    // D layout: VGPR v, lane -> row = v + 8*half (batch), col = l16 (feature).
    // Bias needs input[b0 + v + 8*half] per accumulator register.
    float inrow[8];
    #pragma unroll
    for (int v = 0; v < 8; ++v)
        inrow[v] = inp[b0 + v + 8 * half];

    // ---- 4 output-feature tiles of 16 columns each ----
    #pragma unroll
    for (int m0 = 0; m0 < 64; m0 += 16) {
        // Fold the rank-1 bias into the initial accumulator: the WMMA +C
        // chain then carries it through all 16 K-steps for free.
        float bt = Bt[m0 + l16];
        v8f acc;
        #pragma unroll
        for (int v = 0; v < 8; ++v)
            acc[v] = bt * inrow[v];

        // B operand: B_mat[k, m] = A[t][m0 + m, 4*kc + k]; lane holds N = l16,
        // K = vgpr + 2*half -> contiguous float2 at sAt[(m0+l16)*S + 4*kc + 2*half].
        const float* sb = sAt + (m0 + l16) * LDS_STRIDE + 2 * half;
        #pragma unroll
        for (int kc = 0; kc < 16; ++kc) {
            v2f bfrag = *(const v2f*)(sb + 4 * kc);
            acc = __builtin_amdgcn_wmma_f32_16x16x4_f32(
                /*neg_a=*/false, afrag[kc],
                /*neg_b=*/false, bfrag,
                /*c_mod=*/(short)0, acc,
                /*reuse_a=*/false, /*reuse_b=*/false);
        }

        // Store: per VGPR, each half-wave writes a contiguous 64 B run.
        #pragma unroll
        for (int v = 0; v < 8; ++v)
            out[(b0 + v + 8 * half) * 64 + (m0 + l16)] = acc[v];
    }
}

extern "C" void kernel_launch(void* const* d_in, const int* in_sizes, int n_in,
                              void* d_out, int out_size, void* d_ws, size_t ws_size,
                              hipStream_t stream) {
    const float* inp   = (const float*)d_in[0];  // input  (BATCH,1,1)
    const float* c_t   = (const float*)d_in[1];  // c_t    (BATCH,1,64)
    const float* Afull = (const float*)d_in[2];  // A      (1024,64,64)
    const float* Bfull = (const float*)d_in[3];  // B      (1024,64,1)
    const int*   t_ptr = (const int*)d_in[4];    // t scalar
    float*       out   = (float*)d_out;          // (BATCH,1,64)

    const int batch  = in_sizes[0];              // 524288
    const int blocks = batch / 128;              // 128 batch rows per block

    hippo_legs_wmma_kernel<<<dim3(blocks), dim3(256), 0, stream>>>(
        inp, c_t, Afull, Bfull, t_ptr, out);
}